// PairwiseRankingLoss_86646670230244
// MI455X (gfx1250) — compile-verified
//
#include <hip/hip_runtime.h>

typedef __attribute__((ext_vector_type(2))) float v2f;
typedef __attribute__((ext_vector_type(8))) float v8f;

#define NROWS   4096
#define DDIM    1024
#define MARGIN_F 0.2f
#define BM      256   // block tile rows (M)
#define BN      128   // block tile cols (N)
#define KC      32    // K chunk
#define LSTR    36    // LDS row stride (floats): rows 16B-aligned, conflict-free frag reads

__global__ void init_acc_kernel(double* acc) { acc[0] = 0.0; }

__global__ void finish_kernel(const double* __restrict__ acc, float* __restrict__ out) {
    out[0] = (float)acc[0];
}

// diag_a[j] = dot(artist[j], tart[j]); diag_g[j] = dot(genre[j], genre[j])
__global__ __launch_bounds__(256) void diag_kernel(const float* __restrict__ artist,
                                                   const float* __restrict__ genre,
                                                   const float* __restrict__ tart,
                                                   float* __restrict__ dA,
                                                   float* __restrict__ dG) {
    const int j = blockIdx.x;
    const int t = threadIdx.x;
    const float* ar = artist + (size_t)j * DDIM;
    const float* ta = tart   + (size_t)j * DDIM;
    const float* ge = genre  + (size_t)j * DDIM;
    float sa = 0.f, sg = 0.f;
    #pragma unroll
    for (int s = 0; s < DDIM / 256; s++) {
        int k = t + s * 256;
        sa += ar[k] * ta[k];
        float g = ge[k];
        sg += g * g;
    }
    __shared__ float rA[256], rG[256];
    rA[t] = sa; rG[t] = sg;
    __syncthreads();
    for (int s = 128; s > 0; s >>= 1) {
        if (t < s) { rA[t] += rA[t + s]; rG[t] += rG[t + s]; }
        __syncthreads();
    }
    if (t == 0) { dA[j] = rA[0]; dG[j] = rG[0]; }
}

// score = A @ B^T (256x128 block tile), then sum(scale * relu(MARGIN - diag[j] + s), i != j)
// 8 waves arranged 4(M) x 2(N); each wave owns a 64x64 subtile = 4x4 v8f accumulators.
// Per k-step: 4 A-frags + 4 B-frags (8 ds_load_b64) feed 16 v_wmma_f32_16x16x4_f32.
__global__ __launch_bounds__(256) void gemm_cost_kernel(const float* __restrict__ A,
                                                        const float* __restrict__ B,
                                                        const float* __restrict__ diag,
                                                        float scale,
                                                        double* __restrict__ acc_out) {
    __shared__ float As[BM * LSTR];   // 256x32 staged A rows
    __shared__ float Bs[BN * LSTR];   // 128x32 staged B(=T) rows
    __shared__ float red[256];

    const int tid  = threadIdx.x;
    const int wave = tid >> 5;      // 0..7 (wave32)
    const int lane = tid & 31;
    const int half = lane >> 4;     // 0/1
    const int l16  = lane & 15;

    const int wm = wave >> 1;       // 0..3 : 64-row band
    const int wn = wave & 1;        // 0..1 : 64-col band

    const int i0 = blockIdx.y * BM;
    const int j0 = blockIdx.x * BN;

    v8f acc[4][4];
    v8f zero = {};
    #pragma unroll
    for (int mi = 0; mi < 4; mi++)
        #pragma unroll
        for (int ni = 0; ni < 4; ni++)
            acc[mi][ni] = zero;

    for (int k0 = 0; k0 < DDIM; k0 += KC) {
        // Stage A[i0:+256, k0:+32]: 2048 float4 slots -> 8 per thread
        #pragma unroll
        for (int t = 0; t < 8; t++) {
            int f   = tid + t * 256;
            int row = f >> 3;              // 0..255
            int c4  = (f & 7) << 2;        // 0,4,...,28
            float4 av = *(const float4*)(A + (size_t)(i0 + row) * DDIM + k0 + c4);
            *(float4*)&As[row * LSTR + c4] = av;
        }
        // Stage B[j0:+128, k0:+32]: 1024 float4 slots -> 4 per thread
        #pragma unroll
        for (int t = 0; t < 4; t++) {
            int f   = tid + t * 256;
            int row = f >> 3;              // 0..127
            int c4  = (f & 7) << 2;
            float4 bv = *(const float4*)(B + (size_t)(j0 + row) * DDIM + k0 + c4);
            *(float4*)&Bs[row * LSTR + c4] = bv;
        }
        __syncthreads();

        #pragma unroll
        for (int kk = 0; kk < KC; kk += 4) {
            v2f afr[4], bfr[4];
            #pragma unroll
            for (int mi = 0; mi < 4; mi++) {
                int r = wm * 64 + mi * 16 + l16;          // A row (M); lane half picks K pair
                afr[mi] = *(const v2f*)&As[r * LSTR + kk + 2 * half];
            }
            #pragma unroll
            for (int ni = 0; ni < 4; ni++) {
                int r = wn * 64 + ni * 16 + l16;          // B(=T) row (N); half picks K pair
                bfr[ni] = *(const v2f*)&Bs[r * LSTR + kk + 2 * half];
            }
            #pragma unroll
            for (int mi = 0; mi < 4; mi++)
                #pragma unroll
                for (int ni = 0; ni < 4; ni++)
                    acc[mi][ni] = __builtin_amdgcn_wmma_f32_16x16x4_f32(
                        false, afr[mi], false, bfr[ni],
                        (short)0, acc[mi][ni], false, false);
        }
        __syncthreads();
    }

    // Epilogue: cost = relu(MARGIN - diag[j] + score), zero diagonal, local sum
    float lsum = 0.f;
    #pragma unroll
    for (int ni = 0; ni < 4; ni++) {
        int j = j0 + wn * 64 + ni * 16 + l16;   // lane = column N
        float base = MARGIN_F - diag[j];
        #pragma unroll
        for (int mi = 0; mi < 4; mi++) {
            #pragma unroll
            for (int vc = 0; vc < 8; vc++) {
                int i = i0 + wm * 64 + mi * 16 + half * 8 + vc;  // C/D: VGPR vc = row vc+8*half
                float c = fmaxf(base + acc[mi][ni][vc], 0.f);
                if (i == j) c = 0.f;
                lsum += c;
            }
        }
    }

    red[tid] = lsum * scale;
    __syncthreads();
    for (int s = 128; s > 0; s >>= 1) {
        if (tid < s) red[tid] += red[tid + s];
        __syncthreads();
    }
    if (tid == 0) atomicAdd(acc_out, (double)red[0]);
}

extern "C" void kernel_launch(void* const* d_in, const int* in_sizes, int n_in,
                              void* d_out, int out_size, void* d_ws, size_t ws_size,
                              hipStream_t stream) {
    // setup_inputs order: vgg, artist, genre, target_vgg, target_art, target_gen
    const float* artist = (const float*)d_in[1];
    const float* genre  = (const float*)d_in[2];
    const float* tart   = (const float*)d_in[4];
    float* out = (float*)d_out;

    double* acc = (double*)d_ws;                  // 8B accumulator at ws[0]
    float*  dA  = (float*)d_ws + 4;               // diag_a[4096] at byte offset 16
    float*  dG  = dA + NROWS;                     // diag_g[4096]

    init_acc_kernel<<<1, 1, 0, stream>>>(acc);
    diag_kernel<<<NROWS, 256, 0, stream>>>(artist, genre, tart, dA, dG);

    dim3 grid(NROWS / BN, NROWS / BM);
    // cost_a counted twice in the reference (cost_vgg == cost_artist)
    gemm_cost_kernel<<<grid, 256, 0, stream>>>(artist, tart, dA, 2.0f, acc);
    gemm_cost_kernel<<<grid, 256, 0, stream>>>(genre, genre, dG, 1.0f, acc);

    finish_kernel<<<1, 1, 0, stream>>>(acc, out);
}